// GHMC_Loss_90546500534448
// MI455X (gfx1250) — compile-verified
//
#include <hip/hip_runtime.h>
#include <hip/hip_bf16.h>

// GHM-C loss, single pass over 128 MB (HBM-bound, ~5.5us floor at 23.3 TB/s).
// mean(ce * w) = (1/N) * sum_b beta[b] * sum_{i in bin b} ce_i
// -> one streaming pass accumulating per-bin {count, ce_sum}; tiny finalize.

#define BINS  10
#define BLOCK 256
#define MAXBLK 2048

// Per-element update into the per-thread LDS histogram slice.
// Layout s_hist[bin*BLOCK + tid] (float2): lane tid uses LDS banks
// (2*tid, 2*tid+1) mod 64 -> conflict-free b64 RMW for any bin value.
__device__ __forceinline__ void ghm_elem(float xe, int te, int tid, float2* s_hist) {
    // g = |sigmoid(x) - t| :  t==0 -> sigmoid(x) = 1/(1+e^-x);  t==1 -> 1/(1+e^x)
    float z = te ? xe : -xe;
    float g = __builtin_amdgcn_rcpf(1.0f + __expf(z));
    int bin = (int)(g * 9.9999f);                 // BINS - 0.0001
    bin = bin < 0 ? 0 : (bin > BINS - 1 ? BINS - 1 : bin);
    // ce = logaddexp(0, x) - x*t  (stable softplus)
    float ce = fmaxf(xe, 0.0f) + __logf(1.0f + __expf(-fabsf(xe))) - (te ? xe : 0.0f);
    int idx = bin * BLOCK + tid;
    float2 v = s_hist[idx];
    v.x += 1.0f;
    v.y += ce;
    s_hist[idx] = v;
}

__global__ __launch_bounds__(BLOCK) void ghm_partial(const float* __restrict__ x,
                                                     const int* __restrict__ tgt,
                                                     float2* __restrict__ ws,   // [gridDim.x * BINS]
                                                     int N) {
    __shared__ float2 s_hist[BINS * BLOCK];       // 20 KB
    const int tid = threadIdx.x;

#pragma unroll
    for (int b = 0; b < BINS; ++b)
        s_hist[b * BLOCK + tid] = make_float2(0.0f, 0.0f);
    __syncthreads();

    const int N4 = N >> 2;
    const float4* __restrict__ xv = (const float4*)x;
    const int4*   __restrict__ tv = (const int4*)tgt;
    const int stride = (int)gridDim.x * BLOCK;

    for (int i = (int)blockIdx.x * BLOCK + tid; i < N4; i += stride) {
        // gfx1250 global_prefetch_b8 for the next tile (speculative, drop-on-fault)
        __builtin_prefetch(xv + i + stride, 0, 0);
        __builtin_prefetch(tv + i + stride, 0, 0);

        float4 xx = xv[i];                         // global_load_b128
        int4   tt = tv[i];                         // global_load_b128
        float xs[4] = {xx.x, xx.y, xx.z, xx.w};    // static-indexed -> registers
        int   ts[4] = {tt.x, tt.y, tt.z, tt.w};
#pragma unroll
        for (int k = 0; k < 4; ++k)
            ghm_elem(xs[k], ts[k], tid, s_hist);
    }

    // scalar tail (N % 4), handled once by block 0
    const int rem = N & 3;
    if (blockIdx.x == 0 && tid < rem) {
        int idx = (N4 << 2) + tid;
        ghm_elem(x[idx], tgt[idx], tid, s_hist);
    }
    __syncthreads();

    // fixed-order tree reduction across the 256 thread slices (deterministic)
    for (int s = BLOCK / 2; s > 0; s >>= 1) {
        if (tid < s) {
#pragma unroll
            for (int b = 0; b < BINS; ++b) {
                float2 a = s_hist[b * BLOCK + tid];
                float2 c = s_hist[b * BLOCK + tid + s];
                a.x += c.x; a.y += c.y;
                s_hist[b * BLOCK + tid] = a;
            }
        }
        __syncthreads();
    }

    if (tid < BINS)
        ws[blockIdx.x * BINS + tid] = s_hist[tid * BLOCK];
}

__global__ __launch_bounds__(BLOCK) void ghm_final(const float2* __restrict__ ws,
                                                   float* __restrict__ out,
                                                   int nblk, float Nf) {
    __shared__ double sc[BINS * BLOCK];            // 20 KB
    __shared__ double ss[BINS * BLOCK];            // 20 KB
    const int tid = threadIdx.x;

#pragma unroll
    for (int b = 0; b < BINS; ++b) {
        double c = 0.0, s = 0.0;
        for (int i = tid; i < nblk; i += BLOCK) {  // fixed order -> deterministic
            float2 w = ws[i * BINS + b];
            c += (double)w.x;
            s += (double)w.y;
        }
        sc[b * BLOCK + tid] = c;
        ss[b * BLOCK + tid] = s;
    }
    __syncthreads();

    for (int st = BLOCK / 2; st > 0; st >>= 1) {
        if (tid < st) {
#pragma unroll
            for (int b = 0; b < BINS; ++b) {
                sc[b * BLOCK + tid] += sc[b * BLOCK + tid + st];
                ss[b * BLOCK + tid] += ss[b * BLOCK + tid + st];
            }
        }
        __syncthreads();
    }

    if (tid == 0) {
        int nonempty = 0;
#pragma unroll
        for (int b = 0; b < BINS; ++b)
            nonempty += (sc[b * BLOCK] > 0.0) ? 1 : 0;
        double loss = 0.0;
#pragma unroll
        for (int b = 0; b < BINS; ++b) {
            double gd = sc[b * BLOCK] * (double)nonempty;
            if (gd < 1e-6) gd = 1e-6;
            double beta = (double)Nf / gd;         // weight for every element in bin b
            loss += beta * ss[b * BLOCK];
        }
        out[0] = (float)(loss / (double)Nf);
    }
}

extern "C" void kernel_launch(void* const* d_in, const int* in_sizes, int n_in,
                              void* d_out, int out_size, void* d_ws, size_t ws_size,
                              hipStream_t stream) {
    const float* x   = (const float*)d_in[0];
    const int*   tgt = (const int*)d_in[1];
    const int N = in_sizes[0];

    // workspace: nblk * BINS float2 partials (160 KB at nblk=2048)
    int nblk = (int)(ws_size / (BINS * sizeof(float2)));
    if (nblk > MAXBLK) nblk = MAXBLK;
    if (nblk < 1) nblk = 1;
    float2* ws = (float2*)d_ws;

    ghm_partial<<<nblk, BLOCK, 0, stream>>>(x, tgt, ws, N);
    ghm_final<<<1, BLOCK, 0, stream>>>(ws, (float*)d_out, nblk, (float)N);
}